// fast_dtw_20633022890319
// MI455X (gfx1250) — compile-verified
//
#include <hip/hip_runtime.h>
#include <math.h>
#include <stdint.h>

typedef float v2f __attribute__((ext_vector_type(2)));
typedef float v8f __attribute__((ext_vector_type(8)));

#define B_ 128
#define C_ 16
#define F_ 128
#define D_ 512
#define ASTR 516            // padded A row stride (floats): banks (4r+k)%64, conflict-free
#define BSTR 36             // padded B row stride (floats): banks (36r+k)%64, conflict-free
#define KC 32               // K-chunk staged per async round
#define NEGINF (-__builtin_inff())

// ---- CDNA5 async global->LDS copy (ASYNCcnt-tracked), per CDNA5 ISA ----
__device__ __forceinline__ void async_load_b128(uint32_t lds_off, const void* gptr) {
  asm volatile("global_load_async_to_lds_b128 %0, %1, off"
               :: "v"(lds_off), "v"((uint64_t)(uintptr_t)gptr)
               : "memory");
}
__device__ __forceinline__ void wait_asynccnt0() {
  asm volatile("s_wait_asynccnt 0x0" ::: "memory");
}
__device__ __forceinline__ void wait_dscnt0() {
  asm volatile("s_wait_dscnt 0x0" ::: "memory");
}

// ---------------- prep: normalize c_feats rows into workspace ----------------
__global__ __launch_bounds__(256) void norm_c_kernel(const float* __restrict__ c_feats,
                                                     float* __restrict__ xn) {
  const int wave = threadIdx.x >> 5, lane = threadIdx.x & 31;
  const int row = blockIdx.x * 8 + wave;              // 128*16 = 2048 rows
  const float* src = c_feats + (size_t)row * D_;
  float s = 0.f;
  for (int k = lane; k < D_; k += 32) { float v = src[k]; s += v * v; }
  for (int off = 16; off; off >>= 1) s += __shfl_xor(s, off, 32);
  const float inv = rsqrtf(s);
  float* dst = xn + (size_t)row * D_;
  for (int k = lane; k < D_; k += 32) dst[k] = src[k] * inv;
}

// ---------------- prep: inverse norms of f_feats rows ----------------
__global__ __launch_bounds__(256) void norm_f_kernel(const float* __restrict__ f_feats,
                                                     float* __restrict__ inv_f) {
  const int wave = threadIdx.x >> 5, lane = threadIdx.x & 31;
  const int row = blockIdx.x * 8 + wave;              // 128*128 = 16384 rows
  const float* src = f_feats + (size_t)row * D_;
  float s = 0.f;
  for (int k = lane; k < D_; k += 32) { float v = src[k]; s += v * v; }
  for (int off = 16; off; off >>= 1) s += __shfl_xor(s, off, 32);
  if (lane == 0) inv_f[row] = rsqrtf(s);
}

// ---------------- main: one workgroup per (m,n) pair ----------------
__global__ __launch_bounds__(256)
void dtw_pair_kernel(const float* __restrict__ xn, const float* __restrict__ f_feats,
                     const float* __restrict__ inv_f, float* __restrict__ out) {
  __shared__ __align__(16) float As[C_ * ASTR];        // 33.0 KB : xn[m], padded rows
  __shared__ __align__(16) float BsF[8 * C_ * BSTR];   // 18.4 KB : per-wave B chunk (reused later)
  __shared__ float Slds[C_ * F_];                      //  8.2 KB : score tile
  __shared__ float Dprev[F_ + 1];                      // DP previous row D[i-1][0..128]

  // after GEMM the B staging buffer is dead: alias mask (8 KB) + moves (2 KB) into it
  float* maskLds = BsF;                                      // 2048 floats
  unsigned char* mvLds = (unsigned char*)(BsF + C_ * F_);    // 2048 bytes

  const int pair = blockIdx.x;
  const int m = pair >> 7;
  const int n = pair & 127;
  const int tid = threadIdx.x;
  const int wave = tid >> 5;
  const int lane = tid & 31;

  // ---- stage A = xn[m] (16 x 512 -> 16 x ASTR) via async b128, coalesced ----
  {
    const uint32_t as_base = (uint32_t)(uintptr_t)(const void*)As;
    const float* srcA = xn + (size_t)m * (C_ * D_);
#pragma unroll
    for (int it = 0; it < 8; ++it) {
      const int idx = it * 256 + tid;        // 0..2047 float4 units
      const int row = idx >> 7;              // 128 float4 per row
      const int kq = (idx & 127) << 2;
      async_load_b128(as_base + (uint32_t)(row * ASTR + kq) * 4u,
                      srcA + row * D_ + kq);
    }
  }
  wait_asynccnt0();
  __syncthreads();

  // ---- GEMM: wave w computes S[:, 16w..16w+15] with fp32 WMMA (K=4) ----
  // A 16x4 layout: lanes 0-15 hold M=lane,K={k,k+1}; lanes 16-31 M=lane-16,K={k+2,k+3}
  // B 4x16 layout: lanes 0-15 hold N=lane,K={k,k+1}; lanes 16-31 N=lane-16,K={k+2,k+3}
  const int fbase = wave * 16;
  const int r15 = lane & 15;
  const int col = fbase + r15;
  const int khalf = (lane >> 4) << 1;                  // 0 or 2
  float* Bwave = BsF + wave * (C_ * BSTR);
  const uint32_t bs_base = (uint32_t)(uintptr_t)(const void*)Bwave;
  const float* Aw = As + r15 * ASTR + khalf;
  const float* Bw = Bwave + r15 * BSTR + khalf;

  v8f acc = {0.f, 0.f, 0.f, 0.f, 0.f, 0.f, 0.f, 0.f};
  for (int kb = 0; kb < D_; kb += KC) {
    // prior chunk's ds_loads must have landed in VGPRs before overwrite
    wait_dscnt0();
    // stage this wave's 16 x KC B chunk: coalesced 128B runs from L2
#pragma unroll
    for (int t = 0; t < (C_ * KC) / (32 * 4); ++t) {   // 4 issues/lane
      const int idx = t * 32 + lane;                   // float4 units, 8 per row
      const int row = idx >> 3;
      const int kq = (idx & 7) << 2;
      async_load_b128(bs_base + (uint32_t)(row * BSTR + kq) * 4u,
                      f_feats + ((size_t)(n * F_ + fbase + row)) * D_ + kb + kq);
    }
    wait_asynccnt0();
#pragma unroll
    for (int kk = 0; kk < KC; kk += 4) {
      v2f a = *(const v2f*)(Aw + kb + kk);             // ds_load_b64, bank-conflict-free
      v2f b = *(const v2f*)(Bw + kk);                  // ds_load_b64, bank-conflict-free
      acc = __builtin_amdgcn_wmma_f32_16x16x4_f32(false, a, false, b,
                                                  (short)0, acc, false, false);
    }
  }

  // ---- scale (cosine where f<=c) and park S in LDS ----
  // C/D layout: VGPR j -> row M = j + 8*(lane>=16); col N = (lane&15)+fbase
  {
    const float invf = inv_f[n * F_ + col];
    const int rbase = (lane >> 4) ? 8 : 0;
#pragma unroll
    for (int j = 0; j < 8; ++j) {
      const int c = rbase + j;
      float v = acc[j];
      if (col <= c) v *= invf;
      Slds[c * F_ + col] = v;
    }
  }

  // ---- init DP border: D[0][0]=0, D[0][1..128]=-inf ----
  if (wave == 0) {
#pragma unroll
    for (int t = 0; t < 4; ++t) Dprev[4 * lane + t + 1] = NEGINF;
    if (lane == 0) Dprev[0] = 0.f;
  }
  __syncthreads();

  // ---- DP rows: row[j] = Ccum[j] + cummax_{k<=j}(max(prev[k],prev[k+1]) - Ccum[k-1]) ----
  for (int i = 1; i <= C_; ++i) {
    if (wave == 0) {
      const float* srow = Slds + (i - 1) * F_;
      const int j0 = 4 * lane;
      const float s0 = srow[j0], s1 = srow[j0 + 1], s2 = srow[j0 + 2], s3 = srow[j0 + 3];
      const float p0 = Dprev[j0],     p1 = Dprev[j0 + 1], p2 = Dprev[j0 + 2];
      const float p3 = Dprev[j0 + 3], p4 = Dprev[j0 + 4];

      // inclusive cumsum of S row (4/lane local + wave scan)
      const float cs0 = s0, cs1 = cs0 + s1, cs2 = cs1 + s2, cs3 = cs2 + s3;
      float incl = cs3;
      for (int off = 1; off < 32; off <<= 1) {
        float v = __shfl_up(incl, off, 32);
        if (lane >= off) incl += v;
      }
      float esum = __shfl_up(incl, 1, 32);
      if (lane == 0) esum = 0.f;
      const float Cc0 = esum + cs0, Cc1 = esum + cs1, Cc2 = esum + cs2, Cc3 = esum + cs3;

      // G[k] = max(prev[k],prev[k+1]) - Ccum[k-1];  Ccum[k-1] == Ccum[k]-s[k]
      const float g0 = fmaxf(p0, p1) - (Cc0 - s0);
      const float g1 = fmaxf(p1, p2) - (Cc1 - s1);
      const float g2 = fmaxf(p2, p3) - (Cc2 - s2);
      const float g3 = fmaxf(p3, p4) - (Cc3 - s3);
      const float m0 = g0, m1 = fmaxf(m0, g1), m2 = fmaxf(m1, g2), m3 = fmaxf(m2, g3);
      float incm = m3;
      for (int off = 1; off < 32; off <<= 1) {
        float v = __shfl_up(incm, off, 32);
        if (lane >= off) incm = fmaxf(incm, v);
      }
      float emax = __shfl_up(incm, 1, 32);
      if (lane == 0) emax = NEGINF;
      const float r0 = Cc0 + fmaxf(emax, m0);
      const float r1 = Cc1 + fmaxf(emax, m1);
      const float r2 = Cc2 + fmaxf(emax, m2);
      const float r3 = Cc3 + fmaxf(emax, m3);

      // moves: argmax(up, left, diag), first-max-wins (up > left > diag)
      float leftNbr = __shfl_up(r3, 1, 32);
      if (lane == 0) leftNbr = NEGINF;                 // D[i][0] = -inf
      const float ups[4] = {p1, p2, p3, p4};
      const float dgs[4] = {p0, p1, p2, p3};
      const float lfs[4] = {leftNbr, r0, r1, r2};
#pragma unroll
      for (int t = 0; t < 4; ++t) {
        const float up = ups[t], lf = lfs[t], dg = dgs[t];
        unsigned char mm = (up >= lf && up >= dg) ? (unsigned char)0
                          : ((lf >= dg) ? (unsigned char)1 : (unsigned char)2);
        mvLds[(i - 1) * F_ + j0 + t] = mm;
      }

      // commit new row (reads of Dprev above precede these writes in program order)
      Dprev[j0 + 1] = r0; Dprev[j0 + 2] = r1; Dprev[j0 + 3] = r2; Dprev[j0 + 4] = r3;
      if (lane == 0) Dprev[0] = NEGINF;
    }
    __syncthreads();
  }

  // ---- zero mask tile, backtrack, write path ----
  for (int t = tid; t < C_ * F_; t += 256) maskLds[t] = 0.f;
  __syncthreads();

  if (tid == 0) {
    int i = C_, j = F_;
    for (int step = 0; step < C_ + F_; ++step) {
      const bool active = (i > 0) || (j > 0);
      if (active && i > 0 && j > 0) maskLds[(i - 1) * F_ + (j - 1)] = 1.0f;
      const int mm = (i > 0 && j > 0) ? (int)mvLds[(i - 1) * F_ + (j - 1)] : 0;
      const int di = (mm == 1) ? 0 : 1;
      const int dj = (mm == 0) ? 0 : 1;
      if (active) { i -= di; j -= dj; }
    }
  }
  __syncthreads();

  // ---- stream tile to HBM, non-temporal (write-once output, bypass cache) ----
  float* outp = out + (size_t)pair * (C_ * F_);
  for (int t = tid; t < C_ * F_; t += 256)
    __builtin_nontemporal_store(maskLds[t], outp + t);
}

extern "C" void kernel_launch(void* const* d_in, const int* in_sizes, int n_in,
                              void* d_out, int out_size, void* d_ws, size_t ws_size,
                              hipStream_t stream) {
  const float* c_feats = (const float*)d_in[0];   // [128,16,512] fp32
  const float* f_feats = (const float*)d_in[1];   // [128,128,512] fp32
  float* xn    = (float*)d_ws;                    // 128*16*512 floats = 4 MB
  float* inv_f = xn + (size_t)B_ * C_ * D_;       // 16384 floats

  norm_c_kernel<<<(B_ * C_) / 8, 256, 0, stream>>>(c_feats, xn);
  norm_f_kernel<<<(B_ * F_) / 8, 256, 0, stream>>>(f_feats, inv_f);
  dtw_pair_kernel<<<B_ * B_, 256, 0, stream>>>(xn, f_feats, inv_f, (float*)d_out);
}